// Attention_4346506903555
// MI455X (gfx1250) — compile-verified
//
#include <hip/hip_runtime.h>
#include <stdint.h>

typedef __attribute__((ext_vector_type(2))) float v2f;
typedef __attribute__((ext_vector_type(8))) float v8f;
typedef __attribute__((ext_vector_type(4))) unsigned int v4u;
typedef __attribute__((ext_vector_type(4))) int v4i;
typedef __attribute__((ext_vector_type(8))) int v8i;

#define B_      16
#define H_      8
#define D_      128
#define T_      2048
#define BS_     16
#define BPS_    128
#define NSPLIT  16
#define CHUNK   (T_ / NSPLIT)   // 128 tokens per workgroup
#define TILES   (CHUNK / 16)    // 8 tiles of 16 tokens
#define PITCH   132             // dwords per LDS row (16B aligned, 4-bank rotation)
#define WS_STRIDE 132           // floats per partial record: 128 out + m + l + pad

#define SCALE_F 0.08838834764831845f

__device__ __forceinline__ uint32_t lds_off(const void* p) {
  return (uint32_t)(uintptr_t)p;   // low 32 bits of generic LDS pointer = LDS byte offset
}

__device__ __forceinline__ void async_b128(uint32_t ldsoff, const float* gaddr) {
  asm volatile("global_load_async_to_lds_b128 %0, %1, off"
               :: "v"(ldsoff), "v"(gaddr) : "memory");
}

// TDM: one descriptor moves a 16x128 f32 tile (row stride H*D elems) into LDS
// with 4 pad dwords inserted every 128 dwords -> LDS pitch 132 dwords.
__device__ __forceinline__ void tdm_load_tile_16x128(uint32_t ldsbyte, const float* gsrc) {
  const uint64_t ga = (uint64_t)(uintptr_t)gsrc;
  v4u g0;
  g0.x = 1u;                                              // count=1 valid user descriptor
  g0.y = ldsbyte;                                         // lds_addr
  g0.z = (uint32_t)ga;                                    // global_addr[31:0]
  g0.w = (uint32_t)((ga >> 32) & 0x01FFFFFFu) | (2u << 30); // global_addr[56:32] | type=2
  v8i g1;
  g1[0] = (int)((2u << 16)        // data_size = 4B
              | (1u << 20)        // pad_enable
              | (6u << 22)        // pad_interval: 128 dwords
              | (3u << 25));      // pad_amount: 4 dwords
  g1[1] = (int)(128u << 16);      // tensor_dim0 = 128 (low16 at [31:16])
  g1[2] = (int)(16u << 16);       // tensor_dim0 hi | tensor_dim1 = 16
  g1[3] = (int)(128u << 16);      // tensor_dim1 hi | tile_dim0 = 128
  g1[4] = 16;                     // tile_dim1 = 16, tile_dim2 = 0
  g1[5] = (int)(H_ * D_);         // tensor_dim0_stride = 1024 elems
  g1[6] = 0;                      // stride0 hi | tensor_dim1_stride lo
  g1[7] = 0;                      // tensor_dim1_stride hi
  v4i z4 = {0, 0, 0, 0};
  v8i z8 = {0, 0, 0, 0, 0, 0, 0, 0};
  __builtin_amdgcn_tensor_load_to_lds(g0, g1, z4, z4, z8, 0);
}

__global__ __launch_bounds__(32)
void pa_partial_kernel(const float* __restrict__ qg, const float* __restrict__ kn,
                       const float* __restrict__ vn, const float* __restrict__ kc,
                       const float* __restrict__ vc, const int* __restrict__ slot,
                       const int* __restrict__ btab, const int* __restrict__ ctxl,
                       float* __restrict__ ws) {
  __shared__ float ksb[2][16 * PITCH];   // double-buffered K tiles (TDM)
  __shared__ float vsb[2][16 * PITCH];   // double-buffered V tiles (async-LDS)
  __shared__ float qs[D_];
  __shared__ float ps[16];

  const int lane   = threadIdx.x;
  const int bh     = blockIdx.x;
  const int split  = blockIdx.y;
  const int b      = bh / H_;
  const int h      = bh % H_;
  const int ctx    = ctxl[b];
  const int chunk0 = split * CHUNK;
  const int sel    = (lane >> 4) << 1;   // 0 for lanes 0-15, 2 for lanes 16-31
  const int tt     = lane & 15;

  // stage q (128 floats) into LDS, then build the 32 A-operand pairs
  const float* qp = qg + (size_t)bh * D_;
  *(float4*)&qs[lane * 4] = *(const float4*)(qp + lane * 4);

  v2f aq[32];
  #pragma unroll
  for (int c = 0; c < 32; ++c)
    aq[c] = *(const v2f*)&qs[4 * c + sel];   // lanes0-15:{q4c,q4c+1} lanes16-31:{q4c+2,q4c+3}

  const float* knew = kn + (size_t)bh * D_;
  const float* vnew = vn + (size_t)bh * D_;
  const int*   btr  = btab + b * BPS_;

  float m = -1e30f, l = 0.f;
  v8f zero = {0.f,0.f,0.f,0.f,0.f,0.f,0.f,0.f};
  v8f acc[8];
  #pragma unroll
  for (int n = 0; n < 8; ++n) acc[n] = zero;

  // base element offset of a tile's (row 0, head h) in the caches
  auto tile_base = [&](int i) -> size_t {
    const int blk = btr[(chunk0 + i * 16) >> 4];
    return ((size_t)blk * BS_ * H_ + h) * D_;
  };

  // ---- prologue: prefetch tile 0 into buffer 0 ----
  {
    const size_t rb = tile_base(0);
    tdm_load_tile_16x128(lds_off(&ksb[0][0]), kc + rb);
    #pragma unroll
    for (int r = 0; r < 16; ++r)
      async_b128(lds_off(&vsb[0][r * PITCH]) + (uint32_t)(lane * 16),
                 vc + rb + (size_t)r * H_ * D_ + lane * 4);
  }

  for (int i = 0; i < TILES; ++i) {
    const int cur = i & 1, nxt = cur ^ 1;

    // ---- prefetch next tile into the other buffer ----
    const int ip = (i + 1 < TILES) ? i + 1 : i;   // clamped (redundant but safe)
    const size_t rbn = tile_base(ip);
    tdm_load_tile_16x128(lds_off(&ksb[nxt][0]), kc + rbn);
    #pragma unroll
    for (int r = 0; r < 16; ++r)
      async_b128(lds_off(&vsb[nxt][r * PITCH]) + (uint32_t)(lane * 16),
                 vc + rbn + (size_t)r * H_ * D_ + lane * 4);

    // ---- wait for CURRENT tile only (in-order completion per counter) ----
    __builtin_amdgcn_s_wait_tensorcnt(1);                 // K(cur) landed
    asm volatile("s_wait_asynccnt 0x10" ::: "memory");    // V(cur) landed
    asm volatile("" ::: "memory");

    // ---- substitute new-token row (reference scatter) ----
    const int tok0 = chunk0 + i * 16;
    const int rfix = ctx - 1 - tok0;
    if (rfix >= 0 && rfix < 16) {
      *(float4*)&ksb[cur][rfix * PITCH + lane * 4] = *(const float4*)(knew + lane * 4);
      *(float4*)&vsb[cur][rfix * PITCH + lane * 4] = *(const float4*)(vnew + lane * 4);
    }

    const float* ksc = &ksb[cur][0];
    const float* vsc = &vsb[cur][0];

    // ---- scores: C(16x16) = Q(16x4)*K^T(4x16) over 32 dim-chunks ----
    v8f csc = zero;
    #pragma unroll
    for (int c = 0; c < 32; ++c) {
      v2f bk = *(const v2f*)&ksc[tt * PITCH + 4 * c + sel];
      csc = __builtin_amdgcn_wmma_f32_16x16x4_f32(false, aq[c], false, bk,
                                                  (short)0, csc, false, false);
    }

    // every lane holds score of token (lane&15) in csc[0] (all C rows identical)
    const int gt = tok0 + tt;
    const bool valid = gt < ctx;
    float s = valid ? csc[0] * SCALE_F : -3.0e38f;

    float tm = s;
    tm = fmaxf(tm, __shfl_xor(tm, 1));
    tm = fmaxf(tm, __shfl_xor(tm, 2));
    tm = fmaxf(tm, __shfl_xor(tm, 4));
    tm = fmaxf(tm, __shfl_xor(tm, 8));
    const float mnew = fmaxf(m, tm);
    const float resc = __expf(m - mnew);
    const float e    = valid ? __expf(s - mnew) : 0.f;
    float esum = e;
    esum += __shfl_xor(esum, 1);
    esum += __shfl_xor(esum, 2);
    esum += __shfl_xor(esum, 4);
    esum += __shfl_xor(esum, 8);
    l = l * resc + esum;
    m = mnew;
    #pragma unroll
    for (int n = 0; n < 8; ++n) acc[n] *= resc;

    if (lane < 16) ps[lane] = e;

    // ---- PV: acc(16x128) += P(16xtokens) * V(16x128) ----
    #pragma unroll
    for (int tg = 0; tg < 4; ++tg) {
      v2f ap = *(const v2f*)&ps[4 * tg + sel];   // broadcast p across M rows
      const int trow = 4 * tg + sel;
      #pragma unroll
      for (int n = 0; n < 8; ++n) {
        v2f bv;
        bv.x = vsc[(trow + 0) * PITCH + n * 16 + tt];
        bv.y = vsc[(trow + 1) * PITCH + n * 16 + tt];
        acc[n] = __builtin_amdgcn_wmma_f32_16x16x4_f32(false, ap, false, bv,
                                                       (short)0, acc[n], false, false);
      }
    }
  }

  // partial record: 128 unnormalized out dims + m + l
  float* wsp = ws + ((size_t)bh * NSPLIT + split) * WS_STRIDE;
  if (lane < 16) {
    #pragma unroll
    for (int n = 0; n < 8; ++n) wsp[n * 16 + lane] = acc[n][0];
  }
  if (lane == 0) { wsp[128] = m; wsp[129] = l; }
  (void)slot;
}

__global__ __launch_bounds__(128)
void pa_combine_kernel(const float* __restrict__ ws, float* __restrict__ out) {
  const int bh = blockIdx.x;
  const int d  = threadIdx.x;
  const float* wsp = ws + (size_t)bh * NSPLIT * WS_STRIDE;
  float M = -1e30f;
  for (int s2 = 0; s2 < NSPLIT; ++s2)
    M = fmaxf(M, wsp[s2 * WS_STRIDE + 128]);
  float L = 0.f, o = 0.f;
  for (int s2 = 0; s2 < NSPLIT; ++s2) {
    const float w = __expf(wsp[s2 * WS_STRIDE + 128] - M);
    L += wsp[s2 * WS_STRIDE + 129] * w;
    o += wsp[s2 * WS_STRIDE + d] * w;
  }
  out[(size_t)bh * D_ + d] = o / (L > 0.f ? L : 1.f);
}

extern "C" void kernel_launch(void* const* d_in, const int* in_sizes, int n_in,
                              void* d_out, int out_size, void* d_ws, size_t ws_size,
                              hipStream_t stream) {
  const float* q    = (const float*)d_in[0];
  const float* k    = (const float*)d_in[1];
  const float* v    = (const float*)d_in[2];
  const float* kc   = (const float*)d_in[3];
  const float* vc   = (const float*)d_in[4];
  const int*   slot = (const int*)d_in[5];
  const int*   btab = (const int*)d_in[6];
  const int*   ctxl = (const int*)d_in[7];
  float* ws = (float*)d_ws;

  dim3 g1(B_ * H_, NSPLIT);
  pa_partial_kernel<<<g1, 32, 0, stream>>>(q, k, v, kc, vc, slot, btab, ctxl, ws);
  pa_combine_kernel<<<B_ * H_, 128, 0, stream>>>(ws, (float*)d_out);
  (void)in_sizes; (void)n_in; (void)out_size; (void)ws_size;
}